// ShapeletNet_66073776882276
// MI455X (gfx1250) — compile-verified
//
#include <hip/hip_runtime.h>
#include <hip/hip_bf16.h>
#include <math.h>

typedef float v2f __attribute__((ext_vector_type(2)));
typedef float v8f __attribute__((ext_vector_type(8)));

constexpr int B = 32;
constexpr int W = 2048;
constexpr int S = 64;
constexpr int L = 32;
constexpr int O = 10;

constexpr int WAVES_PER_BLOCK = 4;                       // 128 threads
constexpr int TILES_PER_B     = W / 16;                  // 128 window tiles per batch
constexpr int BLOCKS_PER_B    = TILES_PER_B / WAVES_PER_BLOCK; // 32
constexpr unsigned FLT_MAX_BITS = 0x7F7FFFFFu;

// ---- wave32 shuffle helpers (explicit ds_bpermute, no wave-size surprises) ----
__device__ __forceinline__ float lane_bcast_f(float v, int srcLane) {
    return __int_as_float(__builtin_amdgcn_ds_bpermute(srcLane << 2, __float_as_int(v)));
}
__device__ __forceinline__ float lane_xor16_f(float v, int lane) {
    return __int_as_float(__builtin_amdgcn_ds_bpermute((lane ^ 16) << 2, __float_as_int(v)));
}

// ---- kernel 1: init per-(b,s) running-min buffer to +FLT_MAX ----
__global__ void shp_init_min(unsigned int* __restrict__ minbuf, int n) {
    int i = blockIdx.x * blockDim.x + threadIdx.x;
    if (i < n) minbuf[i] = FLT_MAX_BITS;
}

// ---- kernel 2: WMMA distance tiles + min reduction ----
__global__ void __launch_bounds__(128)
shp_min_dist(const float* __restrict__ x,
             const float* __restrict__ shp,
             unsigned int* __restrict__ minbuf) {
    __shared__ __align__(16) float s_shp[S * L];  // holds -2 * shapelet
    __shared__ float s_norm[S];                   // ||shapelet||^2

    const int tid = threadIdx.x;

    // Preload codebook (8 KB) into LDS, pre-scaled by -2; compute ||s||^2.
    if (tid < S) {
        const float* r = shp + tid * L;
        float acc = 0.f;
        #pragma unroll
        for (int i = 0; i < L; ++i) {
            float v = r[i];
            acc += v * v;
            s_shp[tid * L + i] = -2.0f * v;
        }
        s_norm[tid] = acc;
    }
    __syncthreads();

    const int lane = tid & 31;
    const int wave = tid >> 5;
    const int row  = lane & 15;      // A-matrix M for this lane
    const int dlt  = lane >> 4;      // 0: K-pair {0,1}, 1: K-pair {2,3} within chunk
    const int b    = blockIdx.x / BLOCKS_PER_B;
    const int tile = (blockIdx.x % BLOCKS_PER_B) * WAVES_PER_BLOCK + wave;
    const int w0   = tile * 16;

    const float* xrow = x + ((size_t)b * W + (w0 + row)) * L;

    // A fragments for the 16x32 window tile: 8 chunks of 16x4 f32.
    v2f a[8];
    float part = 0.f;
    #pragma unroll
    for (int k = 0; k < 8; ++k) {
        const float* p = xrow + 4 * k + 2 * dlt;     // 8B aligned
        v2f v = *(const v2f*)p;
        a[k] = v;
        part += v.x * v.x + v.y * v.y;
    }
    // Full row norm: lane l and l^16 hold complementary halves of the same row.
    float xnorm = part + lane_xor16_f(part, lane);

    // 4 shapelet groups of 16 -> D = X * (-2 S^T), accumulated in f32.
    #pragma unroll
    for (int g = 0; g < 4; ++g) {
        const int n = (lane & 15) + 16 * g;          // B-matrix N for this lane
        const float* srow = s_shp + n * L;
        v8f c = {};
        #pragma unroll
        for (int k = 0; k < 8; ++k) {
            const float* p = srow + 4 * k + 2 * dlt; // 8B aligned LDS
            v2f bb = *(const v2f*)p;
            c = __builtin_amdgcn_wmma_f32_16x16x4_f32(
                    false, a[k], false, bb, (short)0, c, false, false);
        }
        const float sn = s_norm[n];
        float m = 3.402823466e+38f;
        #pragma unroll
        for (int i = 0; i < 8; ++i) {
            // C VGPR i holds window row (i + 8*dlt); fetch its ||x||^2.
            float xn = lane_bcast_f(xnorm, i + 8 * dlt);
            float d2 = fmaxf(xn + sn + c[i], 0.0f);  // c already = -2 x.s
            m = fminf(m, sqrtf(d2));
        }
        // Combine rows 0-7 (dlt=0 lanes) with rows 8-15 (dlt=1 lanes).
        m = fminf(m, lane_xor16_f(m, lane));
        if (lane < 16)
            atomicMin(&minbuf[b * S + n], __float_as_uint(m)); // bits compare == float compare for m >= 0
    }
}

// ---- kernel 3: tiny classifier + sigmoid ----
__global__ void shp_classify(const float* __restrict__ minbuf,
                             const float* __restrict__ cls_w,
                             const float* __restrict__ cls_b,
                             float* __restrict__ out) {
    int t = threadIdx.x;
    if (t >= B * O) return;
    int b = t / O, o = t % O;
    float acc = cls_b[o];
    #pragma unroll 8
    for (int s = 0; s < S; ++s)
        acc += minbuf[b * S + s] * cls_w[s * O + o];
    out[t] = 1.0f / (1.0f + expf(-acc));
}

extern "C" void kernel_launch(void* const* d_in, const int* in_sizes, int n_in,
                              void* d_out, int out_size, void* d_ws, size_t ws_size,
                              hipStream_t stream) {
    const float* x    = (const float*)d_in[0];
    const float* shp  = (const float*)d_in[1];
    const float* cw   = (const float*)d_in[2];
    const float* cb   = (const float*)d_in[3];
    float* out        = (float*)d_out;
    unsigned int* minbuf = (unsigned int*)d_ws;  // B*S floats (bit-pattern min)

    shp_init_min<<<(B * S + 255) / 256, 256, 0, stream>>>(minbuf, B * S);
    shp_min_dist<<<B * BLOCKS_PER_B, WAVES_PER_BLOCK * 32, 0, stream>>>(x, shp, minbuf);
    shp_classify<<<1, 320, 0, stream>>>((const float*)minbuf, cw, cb, out);
}